// PatchCore_76639396430401
// MI455X (gfx1250) — compile-verified
//
#include <hip/hip_runtime.h>
#include <hip/hip_bf16.h>
#include <float.h>
#include <math.h>

typedef __attribute__((ext_vector_type(2))) float v2f;
typedef __attribute__((ext_vector_type(8))) float v8f;

#define D_DIM 128
#define PC_EPS 1e-12f

// ---------------------------------------------------------------------------
// Kernel 0: per-row sum of squares for embedding (Ne rows) and memory (M rows)
// One wave (32 lanes) per row: float4 per lane covers 128 floats exactly.
// ---------------------------------------------------------------------------
__global__ void __launch_bounds__(256)
pc_norms(const float* __restrict__ emb, const float* __restrict__ mem,
         float* __restrict__ x2, float* __restrict__ y2, int Ne, int M) {
    int wave = (int)((blockIdx.x * blockDim.x + threadIdx.x) >> 5);
    int lane = threadIdx.x & 31;
    int total = Ne + M;
    if (wave >= total) return;
    const float* src = (wave < Ne) ? (emb + (size_t)wave * D_DIM)
                                   : (mem + (size_t)(wave - Ne) * D_DIM);
    float4 v = ((const float4*)src)[lane];
    float ss = v.x * v.x + v.y * v.y + v.z * v.z + v.w * v.w;
#pragma unroll
    for (int off = 16; off; off >>= 1) ss += __shfl_xor(ss, off, 32);
    if (lane == 0) {
        if (wave < Ne) x2[wave] = ss;
        else           y2[wave - Ne] = ss;
    }
}

// ---------------------------------------------------------------------------
// Kernel 1: for each embedding row, min over all M bank rows of
//   d2 = |x|^2 + |y|^2 - 2 x.y     (argmin kept as well)
// 16-row tile per block; 8 waves split the 1024 column tiles; f32 WMMA.
//
// A layout (16x4 f32, per ISA): lane<16 -> row=lane, K={4k,4k+1};
//                               lane>=16 -> row=lane-16, K={4k+2,4k+3}
// B layout mirrors A with N in place of M.
// C layout: lane L, vgpr j -> element (M = j + 8*(L>>4), N = L&15)
// ---------------------------------------------------------------------------
__global__ void __launch_bounds__(256)
pc_minmatch(const float* __restrict__ emb, const float* __restrict__ mem,
            const float* __restrict__ x2, const float* __restrict__ y2,
            float* __restrict__ minD2, int* __restrict__ minIdx, int M) {
    __shared__ float sV[16][128];
    __shared__ int   sI[16][128];

    const int tid  = threadIdx.x;
    const int wave = tid >> 5;         // 0..7
    const int lane = tid & 31;
    const int l15  = lane & 15;
    const int hi   = lane >> 4;        // 0/1
    const int rowBase = blockIdx.x * 16;

    // A tile (16 rows x 128) resident in registers, WMMA layout
    v2f a[32];
    {
        const float* ap = emb + (size_t)(rowBase + l15) * D_DIM + 2 * hi;
#pragma unroll
        for (int k = 0; k < 32; ++k)
            a[k] = *(const v2f*)(ap + 4 * k);
    }
    // |x|^2 for the 8 rows this lane accumulates (C layout M = j + 8*hi)
    float xr[8];
#pragma unroll
    for (int j = 0; j < 8; ++j) xr[j] = x2[rowBase + j + 8 * hi];

    float bestV[8];
    int   bestI[8];
#pragma unroll
    for (int j = 0; j < 8; ++j) { bestV[j] = FLT_MAX; bestI[j] = 0; }

    const int nTiles = M >> 4;
    for (int t = wave; t < nTiles; t += 8) {
        const int colBase = t * 16;
        const float* bp = mem + (size_t)(colBase + l15) * D_DIM + 2 * hi;
        v2f b[32];
#pragma unroll
        for (int k = 0; k < 32; ++k)
            b[k] = *(const v2f*)(bp + 4 * k);

        v8f c = {0.f, 0.f, 0.f, 0.f, 0.f, 0.f, 0.f, 0.f};
#pragma unroll
        for (int k = 0; k < 32; ++k)
            c = __builtin_amdgcn_wmma_f32_16x16x4_f32(
                    false, a[k], false, b[k], (short)0, c, false, false);

        const int col = colBase + l15;
        const float yv = y2[col];
#pragma unroll
        for (int j = 0; j < 8; ++j) {
            float d2 = xr[j] + yv - 2.0f * c[j];
            if (d2 < bestV[j]) { bestV[j] = d2; bestI[j] = col; }
        }
    }

#pragma unroll
    for (int j = 0; j < 8; ++j) {
        sV[j + 8 * hi][wave * 16 + l15] = bestV[j];
        sI[j + 8 * hi][wave * 16 + l15] = bestI[j];
    }
    __syncthreads();

    // 16 rows x 128 candidates -> 16 rows x 16, then final 16->1
    {
        int row = tid >> 4, sub = tid & 15;
        float v = FLT_MAX; int idx = 0x7fffffff;
        for (int s = sub; s < 128; s += 16) {
            float vv = sV[row][s]; int ii = sI[row][s];
            if (vv < v || (vv == v && ii < idx)) { v = vv; idx = ii; }
        }
        __syncthreads();
        sV[row][sub] = v; sI[row][sub] = idx;
    }
    __syncthreads();
    if (tid < 16) {
        float v = FLT_MAX; int idx = 0x7fffffff;
        for (int s = 0; s < 16; ++s) {
            float vv = sV[tid][s]; int ii = sI[tid][s];
            if (vv < v || (vv == v && ii < idx)) { v = vv; idx = ii; }
        }
        minD2[rowBase + tid] = v;
        minIdx[rowBase + tid] = idx;
    }
}

// ---------------------------------------------------------------------------
// Kernel 2: per-batch argmax of patch scores (monotonic in d2); emit the
// winning patch row, its NN bank index, and score = sqrt(clip(d2, EPS)).
// ---------------------------------------------------------------------------
__global__ void __launch_bounds__(256)
pc_argmax(const float* __restrict__ minD2, const int* __restrict__ minIdx,
          int patchesPerBatch, int* __restrict__ bRow, int* __restrict__ bNN,
          float* __restrict__ bScore) {
    __shared__ float sV[256];
    __shared__ int   sI[256];
    int b = blockIdx.x, tid = threadIdx.x;
    const float* d = minD2 + (size_t)b * patchesPerBatch;
    float v = -FLT_MAX; int idx = 0;
    for (int p = tid; p < patchesPerBatch; p += 256) {
        float vv = d[p];
        if (vv > v) { v = vv; idx = p; }
    }
    sV[tid] = v; sI[tid] = idx;
    __syncthreads();
    for (int s = 128; s; s >>= 1) {
        if (tid < s) {
            float v2 = sV[tid + s]; int i2 = sI[tid + s];
            if (v2 > sV[tid] || (v2 == sV[tid] && i2 < sI[tid])) {
                sV[tid] = v2; sI[tid] = i2;
            }
        }
        __syncthreads();
    }
    if (tid == 0) {
        int row = b * patchesPerBatch + sI[0];
        bRow[b]   = row;
        bNN[b]    = minIdx[row];
        bScore[b] = sqrtf(fmaxf(sV[0], PC_EPS));
    }
}

// ---------------------------------------------------------------------------
// Kernel 3: per batch -- d2(nn_sample, bank) into LDS, 9 argmin passes
// (ascending, support[0]==nn itself), 9 final distances, softmax, output.
// ---------------------------------------------------------------------------
__global__ void __launch_bounds__(256)
pc_refine(const float* __restrict__ emb, const float* __restrict__ mem,
          const float* __restrict__ x2, const float* __restrict__ y2,
          const int* __restrict__ bRow, const int* __restrict__ bNN,
          const float* __restrict__ bScore, int M, float* __restrict__ out) {
    extern __shared__ float d2s[];            // M floats (64KB for M=16384)
    __shared__ float rV[256];
    __shared__ int   rI[256];
    __shared__ int   supp[9];
    __shared__ float dists[9];

    int b = blockIdx.x, tid = threadIdx.x;
    int wave = tid >> 5, lane = tid & 31;
    int nn = bNN[b];
    const float4* mem4 = (const float4*)mem;
    float4 nv = mem4[(size_t)nn * 32 + lane];
    float nn2 = y2[nn];

    for (int col = wave; col < M; col += 8) {
        float4 mv = mem4[(size_t)col * 32 + lane];
        float dot = nv.x * mv.x + nv.y * mv.y + nv.z * mv.z + nv.w * mv.w;
#pragma unroll
        for (int off = 16; off; off >>= 1) dot += __shfl_xor(dot, off, 32);
        if (lane == 0) d2s[col] = nn2 + y2[col] - 2.0f * dot;
    }
    __syncthreads();

    for (int k = 0; k < 9; ++k) {
        float v = FLT_MAX; int idx = 0x7fffffff;
        for (int col = tid; col < M; col += 256) {
            float vv = d2s[col];
            if (vv < v) { v = vv; idx = col; }
        }
        rV[tid] = v; rI[tid] = idx;
        __syncthreads();
        for (int s = 128; s; s >>= 1) {
            if (tid < s) {
                float v2 = rV[tid + s]; int i2 = rI[tid + s];
                if (v2 < rV[tid] || (v2 == rV[tid] && i2 < rI[tid])) {
                    rV[tid] = v2; rI[tid] = i2;
                }
            }
            __syncthreads();
        }
        if (tid == 0) { supp[k] = rI[0]; d2s[rI[0]] = FLT_MAX; }
        __syncthreads();
    }

    int prow = bRow[b];
    if (wave == 0) {
        const float4* e4 = (const float4*)(emb + (size_t)prow * D_DIM);
        float4 xv = e4[lane];
        float xx2 = x2[prow];
        for (int k = 0; k < 9; ++k) {
            int s = supp[k];
            float4 sv = mem4[(size_t)s * 32 + lane];
            float dot = xv.x * sv.x + xv.y * sv.y + xv.z * sv.z + xv.w * sv.w;
#pragma unroll
            for (int off = 16; off; off >>= 1) dot += __shfl_xor(dot, off, 32);
            if (lane == 0)
                dists[k] = sqrtf(fmaxf(xx2 + y2[s] - 2.0f * dot, PC_EPS));
        }
    }
    __syncthreads();

    if (tid == 0) {
        float m = dists[0];
        for (int k = 1; k < 9; ++k) m = fmaxf(m, dists[k]);
        float sum = 0.f;
        for (int k = 0; k < 9; ++k) sum += expf(dists[k] - m);
        float w0 = expf(dists[0] - m) / sum;
        out[b] = (1.0f - w0) * bScore[b];
    }
}

// ---------------------------------------------------------------------------
extern "C" void kernel_launch(void* const* d_in, const int* in_sizes, int n_in,
                              void* d_out, int out_size, void* d_ws, size_t ws_size,
                              hipStream_t stream) {
    const float* emb = (const float*)d_in[0];
    const float* mem = (const float*)d_in[1];
    const int Ne = in_sizes[0] / D_DIM;     // 6272
    const int M  = in_sizes[1] / D_DIM;     // 16384
    const int B  = out_size;                // 8
    const int patchesPerBatch = Ne / B;

    char* p = (char*)d_ws;
    float* x2     = (float*)p; p += sizeof(float) * Ne;
    float* y2     = (float*)p; p += sizeof(float) * M;
    float* minD2  = (float*)p; p += sizeof(float) * Ne;
    int*   minIdx = (int*)p;   p += sizeof(int) * Ne;
    int*   bRow   = (int*)p;   p += sizeof(int) * B;
    int*   bNN    = (int*)p;   p += sizeof(int) * B;
    float* bScore = (float*)p; p += sizeof(float) * B;

    {
        int waves  = Ne + M;
        int blocks = (waves + 7) / 8;
        pc_norms<<<blocks, 256, 0, stream>>>(emb, mem, x2, y2, Ne, M);
    }
    pc_minmatch<<<Ne / 16, 256, 0, stream>>>(emb, mem, x2, y2, minD2, minIdx, M);
    pc_argmax<<<B, 256, 0, stream>>>(minD2, minIdx, patchesPerBatch, bRow, bNN, bScore);
    pc_refine<<<B, 256, (size_t)M * sizeof(float), stream>>>(
        emb, mem, x2, y2, bRow, bNN, bScore, M, (float*)d_out);
}